// RelativePositionalAttention_37168646979930
// MI455X (gfx1250) — compile-verified
//
#include <hip/hip_runtime.h>
#include <hip/hip_bf16.h>
#include <math.h>

#define N_NODES 50000
#define N_EDGES 800000
#define EMBED   64
#define NHEADS  8
#define HEAD_DIM 8
#define MAXD    100
#define SCALE_F 0.35355339059327373f   // HD^-0.5 = 8^-0.5

typedef float v2f __attribute__((ext_vector_type(2)));
typedef float v8f __attribute__((ext_vector_type(8)));

// ---------------------------------------------------------------------------
// float atomic max via order-preserving integer trick (valid for -inf init)
// ---------------------------------------------------------------------------
__device__ __forceinline__ void atomicMaxFloat(float* addr, float val) {
    if (val >= 0.0f)
        atomicMax((int*)addr, __float_as_int(val));
    else
        atomicMin((unsigned int*)addr, __float_as_uint(val));
}

// ---------------------------------------------------------------------------
// Init: seg_max = -inf, denom = 0, accum = 0
// ---------------------------------------------------------------------------
__global__ __launch_bounds__(256)
void init_kernel(float* __restrict__ seg_max, float* __restrict__ denom,
                 float* __restrict__ accum) {
    int i = blockIdx.x * blockDim.x + threadIdx.x;
    if (i < N_NODES * NHEADS) {
        seg_max[i] = -INFINITY;
        denom[i]   = 0.0f;
    }
    if (i < N_NODES * EMBED) accum[i] = 0.0f;
}

// ---------------------------------------------------------------------------
// One 16x16 output tile of  Y = X @ W + bias  via fp32 WMMA 16x16x4.
// Inlined with compile-time-distinct pointers so loads stay GLOBAL (not FLAT).
//
// A  16x4 (MxK): lane m=lane&15, comps hold K = 2*hi, 2*hi+1  (hi=lane>>4)
// B  4x16 (KxN): lane n=lane&15, comps hold K rows 2*hi, 2*hi+1
// C/D 16x16:     lane n=lane&15, VGPR v -> row  M = v + 8*hi
// ---------------------------------------------------------------------------
__device__ __forceinline__ void gemm_tile_16x16(
    const v2f a[16],
    const float* __restrict__ W, const float* __restrict__ Bias,
    float* __restrict__ Y, int row0, int ct, int m, int hi)
{
    const float bias = Bias[ct * 16 + m];
    v8f c;
    #pragma unroll
    for (int i = 0; i < 8; ++i) c[i] = bias;

    #pragma unroll
    for (int kk = 0; kk < 16; ++kk) {
        const int krow = kk * 4 + 2 * hi;
        v2f b;
        b.x = W[(size_t)krow * EMBED + ct * 16 + m];
        b.y = W[(size_t)(krow + 1) * EMBED + ct * 16 + m];
        c = __builtin_amdgcn_wmma_f32_16x16x4_f32(
                false, a[kk], false, b, (short)0, c, false, false);
    }

    float* yout = Y + (size_t)row0 * EMBED + ct * 16 + m;
    #pragma unroll
    for (int v = 0; v < 8; ++v) {
        yout[(size_t)(v + 8 * hi) * EMBED] = c[v];
    }
}

// Fused QKV projection: one A-fragment load feeds three weight matrices.
// Block = 128 threads = 4 waves; wave w owns output-column tile [16w,16w+16).
__global__ __launch_bounds__(128)
void gemm_qkv_kernel(const float* __restrict__ X,
                     const float* __restrict__ Wq, const float* __restrict__ Bq, float* __restrict__ Q,
                     const float* __restrict__ Wk, const float* __restrict__ Bk, float* __restrict__ K,
                     const float* __restrict__ Wv, const float* __restrict__ Bv, float* __restrict__ V,
                     int nrows)
{
    const int wave = threadIdx.x >> 5;
    const int lane = threadIdx.x & 31;
    const int m    = lane & 15;
    const int hi   = lane >> 4;
    const int row0 = blockIdx.x * 16;
    if (row0 >= nrows) return;
    const int ct = wave;

    v2f a[16];
    const float* xrow = X + (size_t)(row0 + m) * EMBED;
    #pragma unroll
    for (int kk = 0; kk < 16; ++kk)
        a[kk] = *(const v2f*)(xrow + kk * 4 + 2 * hi);    // 8B-aligned global_load_b64

    gemm_tile_16x16(a, Wq, Bq, Q, row0, ct, m, hi);
    gemm_tile_16x16(a, Wk, Bk, K, row0, ct, m, hi);
    gemm_tile_16x16(a, Wv, Bv, V, row0, ct, m, hi);
}

// Output projection: out = accum @ Wo + bo
__global__ __launch_bounds__(128)
void gemm_out_kernel(const float* __restrict__ X,
                     const float* __restrict__ Wo, const float* __restrict__ Bo,
                     float* __restrict__ Y, int nrows)
{
    const int wave = threadIdx.x >> 5;
    const int lane = threadIdx.x & 31;
    const int m    = lane & 15;
    const int hi   = lane >> 4;
    const int row0 = blockIdx.x * 16;
    if (row0 >= nrows) return;
    const int ct = wave;

    v2f a[16];
    const float* xrow = X + (size_t)(row0 + m) * EMBED;
    #pragma unroll
    for (int kk = 0; kk < 16; ++kk)
        a[kk] = *(const v2f*)(xrow + kk * 4 + 2 * hi);

    gemm_tile_16x16(a, Wo, Bo, Y, row0, ct, m, hi);
}

// ---------------------------------------------------------------------------
// Per-edge distance bins:  clamp((|pos[r]-pos[c]|*10).int, -100, 100)+100
// ---------------------------------------------------------------------------
__global__ __launch_bounds__(256)
void edge_bins_kernel(const int* __restrict__ ei, const float* __restrict__ pos,
                      int* __restrict__ bins) {
    int e = blockIdx.x * blockDim.x + threadIdx.x;
    if (e >= N_EDGES) return;
    const int r = ei[e];
    const int c = ei[N_EDGES + e];
    const float dx = pos[3 * r + 0] - pos[3 * c + 0];
    const float dy = pos[3 * r + 1] - pos[3 * c + 1];
    const float dz = pos[3 * r + 2] - pos[3 * c + 2];
    const float dist = sqrtf(dx * dx + dy * dy + dz * dz);
    int b = (int)(dist * 10.0f);              // trunc toward zero (dist >= 0)
    b = min(max(b, -MAXD), MAXD) + MAXD;
    bins[e] = b;
}

// ---------------------------------------------------------------------------
// Per-(edge,head) scores:  s = scale * q[row] . (k[col] + rel_k[bin])
// plus segment max via float atomics.
// ---------------------------------------------------------------------------
__global__ __launch_bounds__(256)
void edge_score_kernel(const int* __restrict__ ei, const int* __restrict__ bins,
                       const float* __restrict__ Q, const float* __restrict__ K,
                       const float* __restrict__ relk,
                       float* __restrict__ scores, float* __restrict__ seg_max) {
    int idx = blockIdx.x * blockDim.x + threadIdx.x;
    if (idx >= N_EDGES * NHEADS) return;
    const int e = idx >> 3;
    const int h = idx & 7;
    const int r = ei[e];
    const int c = ei[N_EDGES + e];
    const float4* qp = (const float4*)(Q    + (size_t)r       * EMBED + h * HEAD_DIM);
    const float4* kp = (const float4*)(K    + (size_t)c       * EMBED + h * HEAD_DIM);
    const float4* rp = (const float4*)(relk + (size_t)bins[e] * EMBED + h * HEAD_DIM);
    const float4 q0 = qp[0], q1 = qp[1];
    const float4 k0 = kp[0], k1 = kp[1];
    const float4 r0 = rp[0], r1 = rp[1];
    float s = q0.x * (k0.x + r0.x) + q0.y * (k0.y + r0.y)
            + q0.z * (k0.z + r0.z) + q0.w * (k0.w + r0.w)
            + q1.x * (k1.x + r1.x) + q1.y * (k1.y + r1.y)
            + q1.z * (k1.z + r1.z) + q1.w * (k1.w + r1.w);
    s *= SCALE_F;
    scores[idx] = s;
    atomicMaxFloat(&seg_max[r * NHEADS + h], s);
}

// non-finite seg_max (isolated nodes) -> 0, matching the reference
__global__ __launch_bounds__(256)
void segmax_fix_kernel(float* __restrict__ seg_max) {
    int i = blockIdx.x * blockDim.x + threadIdx.x;
    if (i >= N_NODES * NHEADS) return;
    const float v = seg_max[i];
    if (!__builtin_isfinite(v)) seg_max[i] = 0.0f;
}

// ex = exp(s - seg_max[row]); denominators via atomicAdd
__global__ __launch_bounds__(256)
void edge_exp_kernel(const int* __restrict__ ei, float* __restrict__ scores,
                     const float* __restrict__ seg_max, float* __restrict__ denom) {
    int idx = blockIdx.x * blockDim.x + threadIdx.x;
    if (idx >= N_EDGES * NHEADS) return;
    const int e = idx >> 3;
    const int h = idx & 7;
    const int r = ei[e];
    const float ex = expf(scores[idx] - seg_max[r * NHEADS + h]);
    scores[idx] = ex;
    atomicAdd(&denom[r * NHEADS + h], ex);
}

// msg = (ex/denom[row]) * (v[col] + rel_v[bin]); scatter-add into accum[row]
__global__ __launch_bounds__(256)
void edge_msg_kernel(const int* __restrict__ ei, const int* __restrict__ bins,
                     const float* __restrict__ scores, const float* __restrict__ denom,
                     const float* __restrict__ V, const float* __restrict__ relv,
                     float* __restrict__ accum) {
    int idx = blockIdx.x * blockDim.x + threadIdx.x;
    if (idx >= N_EDGES * NHEADS) return;
    const int e = idx >> 3;
    const int h = idx & 7;
    const int r = ei[e];
    const int c = ei[N_EDGES + e];
    const float attn = scores[idx] / denom[r * NHEADS + h];
    const float4* vp = (const float4*)(V    + (size_t)c       * EMBED + h * HEAD_DIM);
    const float4* rp = (const float4*)(relv + (size_t)bins[e] * EMBED + h * HEAD_DIM);
    const float4 v0 = vp[0], v1 = vp[1];
    const float4 r0 = rp[0], r1 = rp[1];
    float* o = accum + (size_t)r * EMBED + h * HEAD_DIM;
    atomicAdd(o + 0, attn * (v0.x + r0.x));
    atomicAdd(o + 1, attn * (v0.y + r0.y));
    atomicAdd(o + 2, attn * (v0.z + r0.z));
    atomicAdd(o + 3, attn * (v0.w + r0.w));
    atomicAdd(o + 4, attn * (v1.x + r1.x));
    atomicAdd(o + 5, attn * (v1.y + r1.y));
    atomicAdd(o + 6, attn * (v1.z + r1.z));
    atomicAdd(o + 7, attn * (v1.w + r1.w));
}

// ---------------------------------------------------------------------------
extern "C" void kernel_launch(void* const* d_in, const int* in_sizes, int n_in,
                              void* d_out, int out_size, void* d_ws, size_t ws_size,
                              hipStream_t stream) {
    (void)in_sizes; (void)n_in; (void)out_size; (void)ws_size;
    const float* x    = (const float*)d_in[0];
    const int*   ei   = (const int*)  d_in[1];
    const float* pos  = (const float*)d_in[2];
    const float* Wq   = (const float*)d_in[3];  const float* bq = (const float*)d_in[4];
    const float* Wk   = (const float*)d_in[5];  const float* bk = (const float*)d_in[6];
    const float* Wv   = (const float*)d_in[7];  const float* bv = (const float*)d_in[8];
    const float* relk = (const float*)d_in[9];
    const float* relv = (const float*)d_in[10];
    const float* Wo   = (const float*)d_in[11]; const float* bo = (const float*)d_in[12];
    float* out = (float*)d_out;

    // workspace layout (floats)
    const size_t NE = (size_t)N_NODES * EMBED;              // 3.2M
    float* ws      = (float*)d_ws;
    float* Q       = ws;
    float* K       = Q + NE;
    float* V       = K + NE;
    float* accum   = V + NE;
    float* scores  = accum + NE;                            // E*H = 6.4M
    float* seg_max = scores + (size_t)N_EDGES * NHEADS;     // N*H
    float* denom   = seg_max + (size_t)N_NODES * NHEADS;    // N*H
    int*   bins    = (int*)(denom + (size_t)N_NODES * NHEADS);

    const int EH = N_EDGES * NHEADS;

    init_kernel<<<(N_NODES * EMBED + 255) / 256, 256, 0, stream>>>(seg_max, denom, accum);

    // fused Q/K/V projection (one A load, three WMMA tile-chains per wave)
    gemm_qkv_kernel<<<N_NODES / 16, 128, 0, stream>>>(
        x, Wq, bq, Q, Wk, bk, K, Wv, bv, V, N_NODES);

    edge_bins_kernel<<<(N_EDGES + 255) / 256, 256, 0, stream>>>(ei, pos, bins);

    edge_score_kernel<<<(EH + 255) / 256, 256, 0, stream>>>(
        ei, bins, Q, K, relk, scores, seg_max);

    segmax_fix_kernel<<<(N_NODES * NHEADS + 255) / 256, 256, 0, stream>>>(seg_max);

    edge_exp_kernel<<<(EH + 255) / 256, 256, 0, stream>>>(ei, scores, seg_max, denom);

    edge_msg_kernel<<<(EH + 255) / 256, 256, 0, stream>>>(
        ei, bins, scores, denom, V, relv, accum);

    // out = accum @ Wo + bo
    gemm_out_kernel<<<N_NODES / 16, 128, 0, stream>>>(accum, Wo, bo, out, N_NODES);
}